// PointConv_SM_36885179138572
// MI455X (gfx1250) — compile-verified
//
#include <hip/hip_runtime.h>

typedef __attribute__((ext_vector_type(2))) float v2f;
typedef __attribute__((ext_vector_type(8))) float v8f;

#define NPTS  65536
#define BATCH 2
#define KNN   16
#define CIN   32
#define COUT  32
#define WSTRIDE 36   // padded channel stride in LDS weight table (16B-aligned rows)

// ---- pass 1: transpose fea (B,32,N) -> feaT (B,N,32) so a gathered neighbor
// column is 128 contiguous bytes (2 cachelines) instead of 32 scattered ones ----
__global__ __launch_bounds__(256) void fea_transpose(
    const float* __restrict__ fea, float* __restrict__ feaT)
{
    const int p = blockIdx.x * blockDim.x + threadIdx.x;   // 0 .. B*N-1 exactly
    const int b = p >> 16;
    const int n = p & (NPTS - 1);
    float v[CIN];
    #pragma unroll
    for (int c = 0; c < CIN; ++c)                          // coalesced reads per c
        v[c] = fea[(long)(b * CIN + c) * NPTS + n];
    float4* dst = (float4*)(feaT + (long)p * CIN);         // 128B contiguous per thread
    #pragma unroll
    for (int q = 0; q < CIN / 4; ++q)
        dst[q] = make_float4(v[4 * q], v[4 * q + 1], v[4 * q + 2], v[4 * q + 3]);
}

__global__ __launch_bounds__(256) void pointconv_wmma_f32(
    const float* __restrict__ rel_xyz,     // (B,3,K,N)
    const float* __restrict__ sample_xyz,  // (B,1,K,N,3)
    const float* __restrict__ feaT,        // (B,N,32) transposed features
    const int*   __restrict__ knn_idx,     // (B,N,K)
    const float* __restrict__ w_1x1,       // (32,35)
    const float* __restrict__ conv_dw,     // (1,32,3,3,3)
    float* __restrict__ out,               // (B,32,N)
    int totalWaves)
{
    // Transposed depthwise-kernel table: ldsT[lin*36 + o], 27*36*4 = 3888 B.
    __shared__ float ldsT[27 * WSTRIDE];
    for (int i = threadIdx.x; i < COUT * 27; i += blockDim.x) {
        const int o   = i / 27;
        const int lin = i - o * 27;
        ldsT[lin * WSTRIDE + o] = conv_dw[i];
    }
    __syncthreads();
    const float4* T4 = (const float4*)ldsT;

    const int lane = threadIdx.x & 31;
    const int m    = lane & 15;            // A row / B column within tile
    const int half = lane >> 4;
    const int waveId = (blockIdx.x * blockDim.x + threadIdx.x) >> 5;

    // ---- A matrix: w_1x1 (32x35, zero-padded to 32x36), resident in VGPRs ----
    // 16x4 f32 A layout: VGPR0 = K=4j+2*half, VGPR1 = K=4j+2*half+1.
    v2f a[2][9];
    #pragma unroll
    for (int t = 0; t < 2; ++t) {
        #pragma unroll
        for (int j = 0; j < 9; ++j) {
            const int c0  = 4 * j + 2 * half;          // <= 34 always
            const int row = m + 16 * t;
            a[t][j].x = w_1x1[row * 35 + c0];
            a[t][j].y = (c0 + 1 < 35) ? w_1x1[row * 35 + c0 + 1] : 0.0f;
        }
    }

    const int k = m;                       // neighbor index == B-matrix column

    for (int p = waveId; p < BATCH * NPTS; p += totalWaves) {   // wave-uniform loop
        const int b = p >> 16;             // p / NPTS
        const int n = p & (NPTS - 1);      // p % NPTS

        // ---- gather B column x[:, k, n]: one contiguous 32-float feature column ----
        const int idx = __builtin_nontemporal_load(&knn_idx[(b * NPTS + n) * KNN + k]);
        const float2* col2 = (const float2*)(feaT + (long)(b * NPTS + idx) * CIN);
        const long relB = (long)(b * 3 * KNN + k) * NPTS + n;

        v2f bv[9];
        #pragma unroll
        for (int j = 0; j < 8; ++j) {      // rows 0..31: 8x b64 off one base address
            const float2 w2 = col2[2 * j + half];      // rows 4j+2h, 4j+2h+1
            bv[j].x = w2.x;
            bv[j].y = w2.y;
        }
        {   // rows 32..35: rel_xyz (3 rows, streamed once) + zero pad
            const int d0 = 2 * half;
            bv[8].x = __builtin_nontemporal_load(&rel_xyz[relB + (long)d0 * KNN * NPTS]);
            bv[8].y = half ? 0.0f
                           : __builtin_nontemporal_load(&rel_xyz[relB + (long)KNN * NPTS]);
        }

        // ---- conv tile (32x16) = w_1x1 (32x36) * x (36x16): 18 fp32 WMMAs,
        //      split even/odd-j accumulators to break the C->C dependence chain ----
        v8f acc0a = {}, acc0b = {}, acc1a = {}, acc1b = {};
        #pragma unroll
        for (int j = 0; j < 9; j += 2) {
            acc0a = __builtin_amdgcn_wmma_f32_16x16x4_f32(
                        false, a[0][j], false, bv[j], (short)0, acc0a, false, false);
            acc1a = __builtin_amdgcn_wmma_f32_16x16x4_f32(
                        false, a[1][j], false, bv[j], (short)0, acc1a, false, false);
        }
        #pragma unroll
        for (int j = 1; j < 9; j += 2) {
            acc0b = __builtin_amdgcn_wmma_f32_16x16x4_f32(
                        false, a[0][j], false, bv[j], (short)0, acc0b, false, false);
            acc1b = __builtin_amdgcn_wmma_f32_16x16x4_f32(
                        false, a[1][j], false, bv[j], (short)0, acc1b, false, false);
        }
        const v8f acc0 = acc0a + acc0b;
        const v8f acc1 = acc1a + acc1b;

        // ---- nearest grid-sample -> depthwise weight index for column k ----
        const long sB = ((long)(b * KNN + k) * NPTS + n) * 3;
        const float sx = __builtin_nontemporal_load(&sample_xyz[sB + 0]);
        const float sy = __builtin_nontemporal_load(&sample_xyz[sB + 1]);
        const float sz = __builtin_nontemporal_load(&sample_xyz[sB + 2]);
        // round((s+1)*1.5 - 0.5) == rint(1.5*s + 1), half-to-even like jnp.round
        const int iw = min(max((int)rintf(fmaf(sx, 1.5f, 1.0f)), 0), 2);
        const int ih = min(max((int)rintf(fmaf(sy, 1.5f, 1.0f)), 0), 2);
        const int id = min(max((int)rintf(fmaf(sz, 1.5f, 1.0f)), 0), 2);
        const int lin = id * 9 + ih * 3 + iw;

        // ---- scale by depthwise weight, pair-reduce over K=16 columns, store ----
        const long outB = (long)(b * COUT) * NPTS + n;
        #pragma unroll
        for (int t = 0; t < 2; ++t) {
            // 2x ds_load_b128: weights for channels (8*half+16*t) .. +7
            const int wbase = lin * (WSTRIDE / 4) + 2 * half + 4 * t;
            const float4 wA = T4[wbase];
            const float4 wB = T4[wbase + 1];

            float pr[8];
            #pragma unroll
            for (int v = 0; v < 8; ++v) {
                const float wv = (v < 4) ? ((const float*)&wA)[v]
                                         : ((const float*)&wB)[v - 4];
                pr[v] = (t ? acc1[v] : acc0[v]) * wv;
            }

            // paired butterfly: 8 regs -> 1, channel ends at lane c = lane&7
            float q[4];
            #pragma unroll
            for (int v = 0; v < 4; ++v) {
                const float keep = (lane & 1) ? pr[2 * v + 1] : pr[2 * v];
                const float send = (lane & 1) ? pr[2 * v]     : pr[2 * v + 1];
                q[v] = keep + __shfl_xor(send, 1, 32);
            }
            float r[2];
            #pragma unroll
            for (int v = 0; v < 2; ++v) {
                const float keep = (lane & 2) ? q[2 * v + 1] : q[2 * v];
                const float send = (lane & 2) ? q[2 * v]     : q[2 * v + 1];
                r[v] = keep + __shfl_xor(send, 2, 32);
            }
            float s;
            {
                const float keep = (lane & 4) ? r[1] : r[0];
                const float send = (lane & 4) ? r[0] : r[1];
                s = keep + __shfl_xor(send, 4, 32);
            }
            s += __shfl_xor(s, 8, 32);     // sum over all 16 columns

            // lane holds channel o = (lane&7) + 8*half + 16*t (dup in lane^8)
            if ((lane & 8) == 0) {
                const int o = (lane & 7) + 8 * half + 16 * t;
                __builtin_nontemporal_store(s, &out[outB + (long)o * NPTS]);
            }
        }
    }
}

extern "C" void kernel_launch(void* const* d_in, const int* in_sizes, int n_in,
                              void* d_out, int out_size, void* d_ws, size_t ws_size,
                              hipStream_t stream) {
    const float* rel_xyz    = (const float*)d_in[0];
    const float* sample_xyz = (const float*)d_in[1];
    const float* fea        = (const float*)d_in[2];
    const int*   knn_idx    = (const int*)  d_in[3];
    const float* w_1x1      = (const float*)d_in[4];
    const float* conv_dw    = (const float*)d_in[5];
    float* out  = (float*)d_out;
    float* feaT = (float*)d_ws;            // BATCH*NPTS*CIN floats = 16.8 MB scratch

    // pass 1: feature transpose (channel-major -> point-major)
    fea_transpose<<<(BATCH * NPTS) / 256, 256, 0, stream>>>(fea, feaT);

    // pass 2: WMMA pointconv
    const int threads = 256;               // 8 wave32 per block
    const int blocks  = 512;               // 4096 waves -> 32 points per wave
    const int totalWaves = blocks * threads / 32;
    pointconv_wmma_f32<<<blocks, threads, 0, stream>>>(
        rel_xyz, sample_xyz, feaT, knn_idx, w_1x1, conv_dw, out, totalWaves);
}